// SplatterBlender_52810917871942
// MI455X (gfx1250) — compile-verified
//
#include <hip/hip_runtime.h>
#include <hip/hip_bf16.h>

// SplatterBlender for MI455X (gfx1250, wave32).
// One thread per output pixel; 16x8 tile per 128-thread block.
// 18x10 halo of {colors,px,zbuf} staged into LDS with CDNA5 async
// global->LDS DMA (GLOBAL_LOAD_ASYNC_TO_LDS_B64/B128, ASYNCcnt).
// LDS layouts chosen so compute-phase reads are ds_load_b64/b128 with
// bank windows striding across all 64 banks.

#define IMG      256
#define KLAY     8
#define TILE_W   16
#define TILE_H   8
#define HALO_W   18
#define HALO_H   10
#define NHALO    (HALO_W * HALO_H)          // 180 halo pixels
#define NTHREADS 128

// LDS float offsets:
//   z   : [p][k]   row stride 12 floats (48 B) -> b128 chunks aligned,
//                  window start bank 12p%64 covers all 64 banks
//   px  : [k][p][2]  (8 B per (k,p))
//   col : [k][p][4]  (16 B per (k,p))
#define ZSTR     12
#define Z_OFF    0                           // 180*12 = 2160 floats
#define PX_OFF   (NHALO * ZSTR)              // 16*180 = 2880 floats
#define COL_OFF  (PX_OFF + 16 * NHALO)       // 32*180 = 5760 floats
#define SMEM_F   (COL_OFF + 32 * NHALO)      // 10800 floats = 43200 B

typedef int v2i __attribute__((ext_vector_type(2)));
typedef int v4i __attribute__((ext_vector_type(4)));
typedef __attribute__((address_space(1))) int gbl_i32;
typedef __attribute__((address_space(3))) int lds_i32;
typedef __attribute__((address_space(1))) v2i gbl_v2i;
typedef __attribute__((address_space(3))) v2i lds_v2i;
typedef __attribute__((address_space(1))) v4i gbl_v4i;
typedef __attribute__((address_space(3))) v4i lds_v4i;

// Async global -> LDS copies (tracked by ASYNCcnt). Low 32 bits of a
// generic pointer into LDS are the LDS byte offset (ISA 10.2).
__device__ static __forceinline__ void async_copy_b64(const float* gsrc,
                                                      float* lds_generic) {
  unsigned lofs = (unsigned)(size_t)lds_generic;
#if __has_builtin(__builtin_amdgcn_global_load_async_to_lds_b64)
  __builtin_amdgcn_global_load_async_to_lds_b64(
      (gbl_v2i*)(size_t)gsrc, (lds_v2i*)lofs, 0, 0);
#else
  asm volatile("global_load_async_to_lds_b64 %0, %1, off"
               :: "v"(lofs), "v"((unsigned long long)(size_t)gsrc)
               : "memory");
#endif
}

__device__ static __forceinline__ void async_copy_b128(const float* gsrc,
                                                       float* lds_generic) {
  unsigned lofs = (unsigned)(size_t)lds_generic;
#if __has_builtin(__builtin_amdgcn_global_load_async_to_lds_b128)
  __builtin_amdgcn_global_load_async_to_lds_b128(
      (gbl_v4i*)(size_t)gsrc, (lds_v4i*)lofs, 0, 0);
#else
  asm volatile("global_load_async_to_lds_b128 %0, %1, off"
               :: "v"(lofs), "v"((unsigned long long)(size_t)gsrc)
               : "memory");
#endif
}

__device__ static __forceinline__ void wait_async_zero() {
#if __has_builtin(__builtin_amdgcn_s_wait_asynccnt)
  __builtin_amdgcn_s_wait_asynccnt(0);
#else
  asm volatile("s_wait_asynccnt 0" ::: "memory");
#endif
}

__global__ __launch_bounds__(NTHREADS) void splatter_blender_kernel(
    const float* __restrict__ colors,   // [N][H][W][K][4]
    const float* __restrict__ pxc,      // [N][H][W][K][2]
    const float* __restrict__ zbuf,     // [N][H][W][K]
    const float* __restrict__ bg,       // [3]
    float* __restrict__ out)            // [N][H][W][4]
{
  __shared__ float smem[SMEM_F];

  const int tid = threadIdx.x;
  const int n  = blockIdx.z;
  const int h0 = blockIdx.y * TILE_H;
  const int w0 = blockIdx.x * TILE_W;

  // ---------------- halo fill: async DMA (in-image) / zero (padding) -------
  // zbuf: 2 x 16B chunks per halo pixel -> b128
  for (int i = tid; i < 2 * NHALO; i += NTHREADS) {
    int p = i >> 1, half = i & 1;
    int hr = p / HALO_W, wc = p - hr * HALO_W;
    int hh = h0 + hr - 1, ww = w0 + wc - 1;
    float* l = &smem[Z_OFF + p * ZSTR + half * 4];
    if ((unsigned)hh < (unsigned)IMG && (unsigned)ww < (unsigned)IMG) {
      size_t pix = (size_t)((n * IMG + hh) * IMG + ww);
      async_copy_b128(zbuf + pix * KLAY + half * 4, l);
    } else {
      *(float4*)l = make_float4(0.f, 0.f, 0.f, 0.f);
    }
  }
  // px: 8 x 8B chunks (one per k) per halo pixel -> b64
  for (int i = tid; i < 8 * NHALO; i += NTHREADS) {
    int p = i >> 3, k = i & 7;
    int hr = p / HALO_W, wc = p - hr * HALO_W;
    int hh = h0 + hr - 1, ww = w0 + wc - 1;
    float* l = &smem[PX_OFF + 2 * (k * NHALO + p)];
    if ((unsigned)hh < (unsigned)IMG && (unsigned)ww < (unsigned)IMG) {
      size_t pix = (size_t)((n * IMG + hh) * IMG + ww);
      async_copy_b64(pxc + pix * (KLAY * 2) + k * 2, l);
    } else {
      *(float2*)l = make_float2(0.f, 0.f);
    }
  }
  // colors: 8 x 16B chunks (one per k) per halo pixel -> b128
  for (int i = tid; i < 8 * NHALO; i += NTHREADS) {
    int p = i >> 3, k = i & 7;
    int hr = p / HALO_W, wc = p - hr * HALO_W;
    int hh = h0 + hr - 1, ww = w0 + wc - 1;
    float* l = &smem[COL_OFF + 4 * (k * NHALO + p)];
    if ((unsigned)hh < (unsigned)IMG && (unsigned)ww < (unsigned)IMG) {
      size_t pix = (size_t)((n * IMG + hh) * IMG + ww);
      async_copy_b128(colors + pix * (KLAY * 4) + k * 4, l);
    } else {
      *(float4*)l = make_float4(0.f, 0.f, 0.f, 0.f);
    }
  }

  wait_async_zero();     // drain ASYNCcnt before making data visible
  __syncthreads();       // (compiler adds DScnt wait for the zero stores)

  // ---------------- per-pixel compute --------------------------------------
  const int lx = tid & (TILE_W - 1);
  const int ly = tid >> 4;
  const int pOwn = (ly + 1) * HALO_W + (lx + 1);

  const float4 za = *(const float4*)&smem[Z_OFF + pOwn * ZSTR];
  const float4 zbv = *(const float4*)&smem[Z_OFF + pOwn * ZSTR + 4];
  const float zown[KLAY] = {za.x, za.y, za.z, za.w,
                            zbv.x, zbv.y, zbv.z, zbv.w};
  const float z0 = zown[0];

  // norm = (1+EPS)/sum_d exp(-|off_d|^2/(2*sigma^2)), sigma=0.5 -> /0.5
  const float NORM =
      1.05f / (1.0f + 4.0f * __expf(-2.0f) + 4.0f * __expf(-4.0f));

  float acc[5][3];
#pragma unroll
  for (int c = 0; c < 5; ++c) {
    acc[c][0] = 0.0f; acc[c][1] = 0.0f; acc[c][2] = 0.0f;
  }

#pragma unroll 1   // keep d rolled: inner k-loops unrolled, acc stays in regs
  for (int d = 0; d < 9; ++d) {
    const int di = d / 3, dj = d % 3;         // OFFSETS[d] = (di-1, dj-1)
    // occlusion neighbor: zp[:, kh:kh+H, kw:kw+W] with d=3*kh+kw
    const int pA = (ly + di) * HALO_W + (lx + dj);
    // splat source:      p[:, d%3:d%3+H, d//3:d//3+W]  (transposed!)
    const int pB = (ly + dj) * HALO_W + (lx + di);
    const float fi = (float)(di - 1);         // x-offset, pairs with px[...,0]
    const float fj = (float)(dj - 1);         // y-offset, pairs with px[...,1]

    // ---- occlusion layer for this d (first-min == jnp.argmin) ----
    const float4 na = *(const float4*)&smem[Z_OFF + pA * ZSTR];
    const float4 nb = *(const float4*)&smem[Z_OFF + pA * ZSTR + 4];
    const float zn[KLAY] = {na.x, na.y, na.z, na.w,
                            nb.x, nb.y, nb.z, nb.w};
    const float zn0 = zn[0];
    float min1 = 3.0e38f, min2 = 3.0e38f;
    int id1 = 0, id2 = 0;
#pragma unroll
    for (int k = 0; k < KLAY; ++k) {
      float a1 = fabsf(zn[k] - z0);
      if (a1 < min1) { min1 = a1; id1 = k; }
      float a2 = fabsf(zn0 - zown[k]);
      if (a2 < min2) { min2 = a2; id2 = k; }
    }
    const int occ = (min2 < min1) ? -id2 : id1;

    // ---- gather splat d from source pixel pB, classify per layer ----
#pragma unroll
    for (int k = 0; k < KLAY; ++k) {
      float2 pv = *(const float2*)&smem[PX_OFF + 2 * (k * NHALO + pB)];
      float4 cv = *(const float4*)&smem[COL_OFF + 4 * (k * NHALO + pB)];
      float tx = floorf(pv.x) - pv.x + 0.5f + fi;
      float ty = floorf(pv.y) - pv.y + 0.5f + fj;
      float g  = __expf(-2.0f * (tx * tx + ty * ty));
      float sw = g * cv.w * NORM;
      float m0 = (occ >  k) ? 1.0f : 0.0f;
      float m1 = (occ == k) ? 1.0f : 0.0f;
      float m2 = (occ <  k) ? 1.0f : 0.0f;
      float v0 = sw * cv.x, v1 = sw * cv.y, v2 = sw * cv.z, v3 = sw * cv.w;
      acc[0][0] = fmaf(v0, m0, acc[0][0]);
      acc[0][1] = fmaf(v0, m1, acc[0][1]);
      acc[0][2] = fmaf(v0, m2, acc[0][2]);
      acc[1][0] = fmaf(v1, m0, acc[1][0]);
      acc[1][1] = fmaf(v1, m1, acc[1][1]);
      acc[1][2] = fmaf(v1, m2, acc[1][2]);
      acc[2][0] = fmaf(v2, m0, acc[2][0]);
      acc[2][1] = fmaf(v2, m1, acc[2][1]);
      acc[2][2] = fmaf(v2, m2, acc[2][2]);
      acc[3][0] = fmaf(v3, m0, acc[3][0]);
      acc[3][1] = fmaf(v3, m1, acc[3][1]);
      acc[3][2] = fmaf(v3, m2, acc[3][2]);
      acc[4][0] = fmaf(sw, m0, acc[4][0]);
      acc[4][1] = fmaf(sw, m1, acc[4][1]);
      acc[4][2] = fmaf(sw, m2, acc[4][2]);
    }
  }

  // ---------------- normalize + back-to-front composite --------------------
  float o0 = bg[0], o1 = bg[1], o2 = bg[2], o3 = 0.0f;
#pragma unroll
  for (int l = 2; l >= 0; --l) {
    float wgt = fmaxf(acc[4][l], 1.0f);
    float n0 = acc[0][l] / wgt;
    float n1 = acc[1][l] / wgt;
    float n2 = acc[2][l] / wgt;
    float n3 = acc[3][l] / wgt;
    float om = 1.0f - n3;
    o0 = n0 + om * o0;
    o1 = n1 + om * o1;
    o2 = n2 + om * o2;
    o3 = n3 + om * o3;
  }

  const int h = h0 + ly, w = w0 + lx;
  float4 r; r.x = o0; r.y = o1; r.z = o2; r.w = o3;
  *(float4*)(out + ((size_t)((n * IMG + h) * IMG + w)) * 4) = r;
}

extern "C" void kernel_launch(void* const* d_in, const int* in_sizes, int n_in,
                              void* d_out, int out_size, void* d_ws,
                              size_t ws_size, hipStream_t stream) {
  (void)in_sizes; (void)n_in; (void)out_size; (void)d_ws; (void)ws_size;
  const float* colors = (const float*)d_in[0];
  const float* pxc    = (const float*)d_in[1];
  const float* zbuf   = (const float*)d_in[2];
  const float* bg     = (const float*)d_in[3];
  float* out = (float*)d_out;

  dim3 grid(IMG / TILE_W, IMG / TILE_H, 2);   // (16, 32, 2)
  dim3 block(NTHREADS, 1, 1);
  splatter_blender_kernel<<<grid, block, 0, stream>>>(colors, pxc, zbuf, bg,
                                                      out);
}